// SSMRec_60550448939205
// MI455X (gfx1250) — compile-verified
//
#include <hip/hip_runtime.h>
#include <hip/hip_bf16.h>
#include <cstdint>

// Problem dims (match reference)
#define BB   64
#define LL   200
#define HH   128
#define SS   64
#define EE   256
#define NBLK 2
#define KK   101

typedef __attribute__((ext_vector_type(2))) float v2f;
typedef __attribute__((ext_vector_type(8))) float v8f;

// ---------------------------------------------------------------------------
// WMMA f32 GEMM: out[M,N] = A[M,K] @ W[K,N] + bias[N]
// MODE 0: plain write to out.
// MODE 1: fused residual epilogue: resid = (resid + val) * mask[row]
//
// Block = 128 threads (4 waves). Each wave owns a 64(M) x 16(N) strip built
// from MT=4 stacked 16x16 accumulators that all share one B fragment per
// K-step -> 4 v_wmma per B load pair, 0.5 strided b32 loads per WMMA.
// Block covers 64 rows x 64 cols. Grid = (M/64, N/64).
//
// V_WMMA_F32_16X16X4_F32 fragments (wave32):
//   A (16x4): lane&15 = M row; lanes>=16 hold K=2,3 halves (contiguous pair).
//   B (4x16): lane&15 = N col; lanes>=16 hold K=2,3 halves (stride-N pair).
//   C/D (16x16): VGPR v -> row v (lanes 0-15) / row v+8 (lanes 16-31).
// ---------------------------------------------------------------------------
#define MT 4

template<int MODE>
__global__ void gemm_wmma_f32(const float* __restrict__ A, const float* __restrict__ W,
                              const float* __restrict__ bias, float* __restrict__ out,
                              int M, int N, int K,
                              float* __restrict__ resid, const float* __restrict__ mask)
{
    const int lane  = threadIdx.x & 31;
    const int wave  = threadIdx.x >> 5;
    const int mtile = blockIdx.x << 6;                 // 64 rows per block
    const int ntile = (blockIdx.y << 6) + (wave << 4); // 16 cols per wave
    const int mrow  = lane & 15;
    const int khalf = (lane >> 4) << 1;                // 0 or 2
    const int ncol  = lane & 15;

    const float* __restrict__ Ap[MT];
#pragma unroll
    for (int mt = 0; mt < MT; ++mt)
        Ap[mt] = A + (size_t)(mtile + mt * 16 + mrow) * K + khalf;
    const float* __restrict__ Wc = W + (size_t)khalf * N + (ntile + ncol);

    __builtin_prefetch(Ap[0], 0, 3);
    __builtin_prefetch(Wc, 0, 3);

    v8f c[MT];
#pragma unroll
    for (int mt = 0; mt < MT; ++mt)
        c[mt] = (v8f){0.f, 0.f, 0.f, 0.f, 0.f, 0.f, 0.f, 0.f};

    for (int k = 0; k < K; k += 4) {
        v2f b;
        b.x = Wc[(size_t)k * N];          // strided pair (K rows of W)
        b.y = Wc[(size_t)(k + 1) * N];
#pragma unroll
        for (int mt = 0; mt < MT; ++mt) {
            v2f a;
            a.x = Ap[mt][k + 0];          // contiguous pair -> b64 load
            a.y = Ap[mt][k + 1];
            c[mt] = __builtin_amdgcn_wmma_f32_16x16x4_f32(
                        /*neg_a=*/false, a, /*neg_b=*/false, b,
                        /*c_mod=*/(short)0, c[mt], /*reuse_a=*/false, /*reuse_b=*/false);
        }
    }

    const int   col   = ntile + ncol;
    const float bv    = bias[col];
    const int   hi8   = (lane >> 4) << 3;              // +8 for upper half-wave

#pragma unroll
    for (int mt = 0; mt < MT; ++mt) {
        const int rbase = mtile + mt * 16 + hi8;
        if (MODE == 0) {
#pragma unroll
            for (int v = 0; v < 8; ++v)
                out[(size_t)(rbase + v) * N + col] = c[mt][v] + bv;
        } else {
#pragma unroll
            for (int v = 0; v < 8; ++v) {
                const int    row = rbase + v;
                const size_t o   = (size_t)row * N + col;
                resid[o] = (resid[o] + c[mt][v] + bv) * mask[row];
            }
        }
    }
}

// ---------------------------------------------------------------------------
// x[b,l,:] = (item_emb[seq[b,l]] + pos_emb[l]) * mask ; cache mask per row.
// grid = B*L, block = H
// ---------------------------------------------------------------------------
__global__ void embed_kernel(const int* __restrict__ seqs, const float* __restrict__ item_emb,
                             const float* __restrict__ pos_emb, float* __restrict__ x,
                             float* __restrict__ maskb)
{
    const int row = blockIdx.x;
    const int l   = row % LL;
    const int tid = threadIdx.x;
    const int it  = seqs[row];
    const float m = (it != 0) ? 1.f : 0.f;
    x[(size_t)row * HH + tid] = (item_emb[(size_t)it * HH + tid] + pos_emb[l * HH + tid]) * m;
    if (tid == 0) maskb[row] = m;
}

// ---------------------------------------------------------------------------
// Row LayerNorm, block = width threads (width <= 256), grid = rows.
// ---------------------------------------------------------------------------
__global__ void ln_rows(const float* __restrict__ in, const float* __restrict__ g,
                        const float* __restrict__ bta, float* __restrict__ out, int width)
{
    __shared__ float sm[256];
    const int row = blockIdx.x, tid = threadIdx.x;
    const float inv = 1.f / (float)width;
    const float v = in[(size_t)row * width + tid];
    sm[tid] = v; __syncthreads();
    for (int o = width >> 1; o > 0; o >>= 1) { if (tid < o) sm[tid] += sm[tid + o]; __syncthreads(); }
    const float mean = sm[0] * inv; __syncthreads();
    const float d = v - mean;
    sm[tid] = d * d; __syncthreads();
    for (int o = width >> 1; o > 0; o >>= 1) { if (tid < o) sm[tid] += sm[tid + o]; __syncthreads(); }
    const float r = rsqrtf(sm[0] * inv + 1e-5f);
    out[(size_t)row * width + tid] = d * r * g[tid] + bta[tid];
}

// ---------------------------------------------------------------------------
// Depthwise centered zero-padded conv1d (k=3, pad=1) + SiLU, plus xsum = sum_e.
// grid = B*L, block = E
// ---------------------------------------------------------------------------
__global__ void conv_silu_kernel(const float* __restrict__ xz, const float* __restrict__ cw,
                                 const float* __restrict__ cb, float* __restrict__ xc,
                                 float* __restrict__ xsum)
{
    __shared__ float sm[EE];
    const int row = blockIdx.x;
    const int l   = row % LL;
    const int e   = threadIdx.x;
    const float x0 = xz[(size_t)row * (2 * EE) + e];
    const float xm = (l > 0)      ? xz[(size_t)(row - 1) * (2 * EE) + e] : 0.f;
    const float xp = (l < LL - 1) ? xz[(size_t)(row + 1) * (2 * EE) + e] : 0.f;
    const float w0 = cw[e * 3 + 0], w1 = cw[e * 3 + 1], w2 = cw[e * 3 + 2];
    float v = xm * w0 + x0 * w1 + xp * w2 + cb[e];
    v = v / (1.f + expf(-v));                 // SiLU
    xc[(size_t)row * EE + e] = v;
    sm[e] = v; __syncthreads();
    for (int o = EE >> 1; o > 0; o >>= 1) { if (e < o) sm[e] += sm[e + o]; __syncthreads(); }
    if (e == 0) xsum[row] = sm[0];
}

// ---------------------------------------------------------------------------
// Collapsed SSM scan. h0==0 and the rank-1 input term is e-independent, so
// all E columns of the state are identical -> keep a single S-vector per b.
//   h_i <- clip( sum_j exp(clip(delta_i*A_j,-10,0)) h_j + dB_i*xsum, +-100 )
//   ysc  = sum_i h_i * C_i
// grid = B, block = S (64 threads, 2 waves); state in LDS.
// ---------------------------------------------------------------------------
__global__ void scan_kernel(const float* __restrict__ ssm, const float* __restrict__ xsum,
                            const float* __restrict__ A_log, float* __restrict__ ysc)
{
    __shared__ float Ash[SS], h[SS], red[SS];
    const int b = blockIdx.x, i = threadIdx.x;
    float al = fminf(fmaxf(A_log[i], -5.f), 5.f);
    Ash[i] = -expf(al);
    h[i]   = 0.f;
    __syncthreads();
    for (int t = 0; t < LL; ++t) {
        const int row = b * LL + t;
        const float* __restrict__ s3 = ssm + (size_t)row * (3 * SS);
        const float d0    = s3[i];
        const float delta = (d0 > 20.f) ? d0 : log1pf(expf(d0));   // softplus
        const float dB    = delta * s3[SS + i];
        const float Cv    = s3[2 * SS + i];
        const float xs    = xsum[row];
        float acc = 0.f;
#pragma unroll 4
        for (int j = 0; j < SS; ++j) {
            const float ad = expf(fminf(fmaxf(delta * Ash[j], -10.f), 0.f));
            acc = fmaf(ad, h[j], acc);
        }
        const float hn = fminf(fmaxf(acc + dB * xs, -100.f), 100.f);
        __syncthreads();                       // all reads of h done
        h[i]   = hn;
        red[i] = hn * Cv;
        __syncthreads();
        for (int o = SS >> 1; o > 0; o >>= 1) { if (i < o) red[i] += red[i + o]; __syncthreads(); }
        if (i == 0) ysc[row] = red[0];
        __syncthreads();
    }
}

// ---------------------------------------------------------------------------
// y = ysc + D*xc ; inner LayerNorm over E ; gate with silu(z). grid=B*L, blk=E
// ---------------------------------------------------------------------------
__global__ void gate_ln_kernel(const float* __restrict__ ysc, const float* __restrict__ xc,
                               const float* __restrict__ xz, const float* __restrict__ D,
                               const float* __restrict__ g, const float* __restrict__ bta,
                               float* __restrict__ yg)
{
    __shared__ float sm[EE];
    const int row = blockIdx.x, e = threadIdx.x;
    const float y = ysc[row] + D[e] * xc[(size_t)row * EE + e];
    sm[e] = y; __syncthreads();
    for (int o = EE >> 1; o > 0; o >>= 1) { if (e < o) sm[e] += sm[e + o]; __syncthreads(); }
    const float mean = sm[0] * (1.f / EE); __syncthreads();
    const float d = y - mean;
    sm[e] = d * d; __syncthreads();
    for (int o = EE >> 1; o > 0; o >>= 1) { if (e < o) sm[e] += sm[e + o]; __syncthreads(); }
    const float r   = rsqrtf(sm[0] * (1.f / EE) + 1e-5f);
    const float lny = d * r * g[e] + bta[e];
    const float z   = xz[(size_t)row * (2 * EE) + EE + e];
    yg[(size_t)row * EE + e] = lny * (z / (1.f + expf(-z)));
}

// ---------------------------------------------------------------------------
// Final LayerNorm on last position only. grid = B, block = H.
// ---------------------------------------------------------------------------
__global__ void final_ln_kernel(const float* __restrict__ x, const float* __restrict__ g,
                                const float* __restrict__ bta, float* __restrict__ xlast)
{
    __shared__ float sm[HH];
    const int b = blockIdx.x, tid = threadIdx.x;
    const size_t row = (size_t)b * LL + (LL - 1);
    const float v = x[row * HH + tid];
    sm[tid] = v; __syncthreads();
    for (int o = HH >> 1; o > 0; o >>= 1) { if (tid < o) sm[tid] += sm[tid + o]; __syncthreads(); }
    const float mean = sm[0] * (1.f / HH); __syncthreads();
    const float d = v - mean;
    sm[tid] = d * d; __syncthreads();
    for (int o = HH >> 1; o > 0; o >>= 1) { if (tid < o) sm[tid] += sm[tid + o]; __syncthreads(); }
    const float r = rsqrtf(sm[0] * (1.f / HH) + 1e-5f);
    xlast[b * HH + tid] = d * r * g[tid] + bta[tid];
}

// ---------------------------------------------------------------------------
// logits[b,k] = dot(xlast[b], item_emb[item_idxs[b,k]]). grid = B, block = H.
// ---------------------------------------------------------------------------
__global__ void logits_kernel(const float* __restrict__ xlast, const int* __restrict__ idxs,
                              const float* __restrict__ item_emb, float* __restrict__ out)
{
    __shared__ float xl[HH];
    const int b = blockIdx.x, tid = threadIdx.x;
    xl[tid] = xlast[b * HH + tid];
    __syncthreads();
    for (int k = tid; k < KK; k += HH) {
        const int it = idxs[b * KK + k];
        const float* __restrict__ emb = item_emb + (size_t)it * HH;
        float acc = 0.f;
#pragma unroll 8
        for (int h = 0; h < HH; ++h) acc = fmaf(xl[h], emb[h], acc);
        out[b * KK + k] = acc;
    }
}

// ---------------------------------------------------------------------------
extern "C" void kernel_launch(void* const* d_in, const int* in_sizes, int n_in,
                              void* d_out, int out_size, void* d_ws, size_t ws_size,
                              hipStream_t stream)
{
    (void)in_sizes; (void)n_in; (void)out_size; (void)ws_size;

    const int*   seqs        = (const int*)d_in[0];
    const int*   idxs        = (const int*)d_in[1];
    const float* item_emb    = (const float*)d_in[2];
    const float* pos_emb     = (const float*)d_in[3];
    const float* blk_ln_g    = (const float*)d_in[4];
    const float* blk_ln_b    = (const float*)d_in[5];
    const float* in_w        = (const float*)d_in[6];
    const float* in_b        = (const float*)d_in[7];
    const float* conv_w      = (const float*)d_in[8];
    const float* conv_b      = (const float*)d_in[9];
    const float* xproj_w     = (const float*)d_in[10];
    const float* xproj_b     = (const float*)d_in[11];
    const float* A_log       = (const float*)d_in[12];
    const float* D_param     = (const float*)d_in[13];
    const float* out_w       = (const float*)d_in[14];
    const float* out_b       = (const float*)d_in[15];
    const float* inner_ln_g  = (const float*)d_in[16];
    const float* inner_ln_b  = (const float*)d_in[17];
    const float* final_ln_g  = (const float*)d_in[18];
    const float* final_ln_b  = (const float*)d_in[19];

    // Workspace carve-out (~76 MB of f32)
    float* wf = (float*)d_ws;
    size_t off = 0;
    float* x     = wf + off; off += (size_t)BB * LL * HH;      // residual stream
    float* u     = wf + off; off += (size_t)BB * LL * HH;      // post-LN activations
    float* xz    = wf + off; off += (size_t)BB * LL * 2 * EE;  // in-proj output
    float* xc    = wf + off; off += (size_t)BB * LL * EE;      // conv+silu
    float* xsum  = wf + off; off += (size_t)BB * LL;           // sum_e xc
    float* ssm   = wf + off; off += (size_t)BB * LL * 3 * SS;  // delta|B|C
    float* ysc   = wf + off; off += (size_t)BB * LL;           // scan scalar output
    float* yg    = wf + off; off += (size_t)BB * LL * EE;      // gated LN output
    float* xlast = wf + off; off += (size_t)BB * HH;           // final LN, last pos
    float* maskb = wf + off; off += (size_t)BB * LL;           // per-row mask

    const int M = BB * LL;   // 12800 rows, divisible by 64

    embed_kernel<<<M, HH, 0, stream>>>(seqs, item_emb, pos_emb, x, maskb);

    for (int blk = 0; blk < NBLK; ++blk) {
        ln_rows<<<M, HH, 0, stream>>>(x, blk_ln_g + blk * HH, blk_ln_b + blk * HH, u, HH);

        // in-proj: [M,128] x [128,512]
        dim3 g1(M / 64, (2 * EE) / 64);
        gemm_wmma_f32<0><<<g1, 128, 0, stream>>>(u, in_w + (size_t)blk * HH * 2 * EE,
                                                 in_b + blk * 2 * EE, xz,
                                                 M, 2 * EE, HH, nullptr, nullptr);

        conv_silu_kernel<<<M, EE, 0, stream>>>(xz, conv_w + blk * EE * 3,
                                               conv_b + blk * EE, xc, xsum);

        // x-proj: [M,256] x [256,192]
        dim3 g2(M / 64, (3 * SS) / 64);
        gemm_wmma_f32<0><<<g2, 128, 0, stream>>>(xc, xproj_w + (size_t)blk * EE * 3 * SS,
                                                 xproj_b + blk * 3 * SS, ssm,
                                                 M, 3 * SS, EE, nullptr, nullptr);

        scan_kernel<<<BB, SS, 0, stream>>>(ssm, xsum, A_log + blk * SS, ysc);

        gate_ln_kernel<<<M, EE, 0, stream>>>(ysc, xc, xz, D_param + blk * EE,
                                             inner_ln_g + blk * EE, inner_ln_b + blk * EE, yg);

        // out-proj: [M,256] x [256,128] with fused (x + xm)*mask residual
        dim3 g3(M / 64, HH / 64);
        gemm_wmma_f32<1><<<g3, 128, 0, stream>>>(yg, out_w + (size_t)blk * EE * HH,
                                                 out_b + blk * HH, nullptr,
                                                 M, HH, EE, x, maskb);
    }

    final_ln_kernel<<<BB, HH, 0, stream>>>(x, final_ln_g, final_ln_b, xlast);
    logits_kernel<<<BB, HH, 0, stream>>>(xlast, idxs, item_emb, (float*)d_out);
}